// FoldNetEncoder_36309653520772
// MI455X (gfx1250) — compile-verified
//
#include <hip/hip_runtime.h>
#include <math.h>

typedef float v2f __attribute__((ext_vector_type(2)));
typedef float v8f __attribute__((ext_vector_type(8)));

#define NPTS 4096
#define BATCH 8
#define PTOT (BATCH * NPTS)
#define KNN_K 16

// ---------------------------------------------------------------------------
// KNN: per query point, keep top-16 by pd = 2*q.c - |q|^2 - |c|^2 (descending),
// stable on ties (strict-> insertion), candidates streamed through LDS tiles.
// ---------------------------------------------------------------------------
__global__ void knn_kernel(const float* __restrict__ pts, int* __restrict__ idx) {
    const int b = blockIdx.y;
    const int q = blockIdx.x * 256 + threadIdx.x;
    const float* px = pts + (size_t)b * 3 * NPTS;
    const float* py = px + NPTS;
    const float* pz = py + NPTS;

    const float qx = px[q], qy = py[q], qz = pz[q];
    const float qq = qx * qx + qy * qy + qz * qz;

    __shared__ float sx[256], sy[256], sz[256];

    float bd[KNN_K];
    int   bi[KNN_K];
#pragma unroll
    for (int j = 0; j < KNN_K; ++j) { bd[j] = -__builtin_inff(); bi[j] = 0; }

    for (int t0 = 0; t0 < NPTS; t0 += 256) {
        __syncthreads();
        sx[threadIdx.x] = px[t0 + threadIdx.x];
        sy[threadIdx.x] = py[t0 + threadIdx.x];
        sz[threadIdx.x] = pz[t0 + threadIdx.x];
        __syncthreads();
        for (int c = 0; c < 256; ++c) {
            const float cx = sx[c], cy = sy[c], cz = sz[c];
            const float cc = cx * cx + cy * cy + cz * cz;
            const float v = 2.f * (qx * cx + qy * cy + qz * cz) - qq - cc;
            if (v > bd[KNN_K - 1]) {
                bd[KNN_K - 1] = v; bi[KNN_K - 1] = t0 + c;
#pragma unroll
                for (int j = KNN_K - 1; j > 0; --j) {
                    if (bd[j] > bd[j - 1]) {
                        float tf = bd[j]; bd[j] = bd[j - 1]; bd[j - 1] = tf;
                        int   ti = bi[j]; bi[j] = bi[j - 1]; bi[j - 1] = ti;
                    }
                }
            }
        }
    }
    int* o = idx + ((size_t)b * NPTS + q) * KNN_K;
#pragma unroll
    for (int j = 0; j < KNN_K; ++j) o[j] = bi[j];
}

// ---------------------------------------------------------------------------
// local_cov: F[0..2] = pts, F[3..11] = outer(nb0, nb1), channel-major [12, P]
// ---------------------------------------------------------------------------
__global__ void local_cov_kernel(const float* __restrict__ pts,
                                 const int* __restrict__ idx,
                                 float* __restrict__ F) {
    const int p = blockIdx.x * blockDim.x + threadIdx.x;
    if (p >= PTOT) return;
    const int b = p >> 12;
    const int n = p & (NPTS - 1);
    const float* pb = pts + (size_t)b * 3 * NPTS;
    const int i0 = idx[(size_t)p * KNN_K + 0];
    const int i1 = idx[(size_t)p * KNN_K + 1];
    const float a0[3] = { pb[i0], pb[NPTS + i0], pb[2 * NPTS + i0] };
    const float a1[3] = { pb[i1], pb[NPTS + i1], pb[2 * NPTS + i1] };
    F[0 * PTOT + p] = pb[n];
    F[1 * PTOT + p] = pb[NPTS + n];
    F[2 * PTOT + p] = pb[2 * NPTS + n];
#pragma unroll
    for (int i = 0; i < 3; ++i)
#pragma unroll
        for (int j = 0; j < 3; ++j)
            F[(size_t)(3 + i * 3 + j) * PTOT + p] = a0[i] * a1[j];
}

// ---------------------------------------------------------------------------
// GEMM Y[Cout,P] = W[Cout,Cin] * X[Cin,P] + bias, optional ReLU.
// One wave computes NT adjacent 16x16 output tiles (NT independent
// accumulator chains -> interleavable WMMAs, A loads amortized NT x).
// ---------------------------------------------------------------------------
template <bool RELU, int NT>
__global__ void gemm_wmma_kernel(const float* __restrict__ W,
                                 const float* __restrict__ X,
                                 const float* __restrict__ bias,
                                 float* __restrict__ Y,
                                 int Cout, int Cin, int P) {
    const int wpb  = blockDim.x >> 5;
    const int wave = blockIdx.x * wpb + (threadIdx.x >> 5);
    const int ngroups = (P >> 4) / NT;
    const int total   = ngroups * (Cout >> 4);
    if (wave >= total) return;                 // wave-uniform: EXEC stays full
    const int mt = wave / ngroups;
    const int ng = wave - mt * ngroups;
    const int m0 = mt << 4, n0 = (ng * NT) << 4;

    const int lane = threadIdx.x & 31;
    const int half = lane >> 4;                // 0: K pair {0,1}; 1: K pair {2,3}
    const int l    = lane & 15;                // A row / B col within tile

    const float* wrow = W + (size_t)(m0 + l) * Cin;
    const float* xcol = X + n0 + l;

    v8f acc[NT];
#pragma unroll
    for (int t = 0; t < NT; ++t) acc[t] = (v8f){};

    for (int k0 = 0; k0 < Cin; k0 += 4) {
        const int ka = k0 + 2 * half;
        v2f a;
        a.x = wrow[ka];
        a.y = wrow[ka + 1];
        const float* xr0 = xcol + (size_t)ka * P;
        const float* xr1 = xcol + (size_t)(ka + 1) * P;
#pragma unroll
        for (int t = 0; t < NT; ++t) {
            v2f b;
            b.x = xr0[t * 16];
            b.y = xr1[t * 16];
            acc[t] = __builtin_amdgcn_wmma_f32_16x16x4_f32(
                false, a, false, b, (short)0, acc[t], false, false);
        }
    }

#pragma unroll
    for (int r = 0; r < 8; ++r) {
        const int row = m0 + r + 8 * half;
        const float bv = bias[row];
        float* yrow = Y + (size_t)row * P + n0 + l;
#pragma unroll
        for (int t = 0; t < NT; ++t) {
            float v = acc[t][r] + bv;
            if (RELU) v = fmaxf(v, 0.f);
            yrow[t * 16] = v;
        }
    }
}

// ---------------------------------------------------------------------------
// local max-pool over k gathered neighbors, channel-major in/out.
// ---------------------------------------------------------------------------
__global__ void maxpool_kernel(const float* __restrict__ X,
                               const int* __restrict__ idx,
                               float* __restrict__ Y, int C) {
    const int p = blockIdx.x * blockDim.x + threadIdx.x;
    if (p >= PTOT) return;
    const int base = (p >> 12) << 12;
    int ind[KNN_K];
    const int* ip = idx + (size_t)p * KNN_K;
#pragma unroll
    for (int j = 0; j < KNN_K; ++j) ind[j] = base + ip[j];
    for (int c = 0; c < C; ++c) {
        const float* row = X + (size_t)c * PTOT;
        float m = -__builtin_inff();
#pragma unroll
        for (int j = 0; j < KNN_K; ++j) m = fmaxf(m, row[ind[j]]);
        Y[(size_t)c * PTOT + p] = m;
    }
}

// ---------------------------------------------------------------------------
// g init: [1024 x 16], cols 0..7 = -inf (max identity), 8..15 = 0 (pad).
// ---------------------------------------------------------------------------
__global__ void init_g_kernel(float* __restrict__ g) {
    const int t = blockIdx.x * blockDim.x + threadIdx.x;
    if (t < 1024 * 16) g[t] = ((t & 15) < 8) ? -__builtin_inff() : 0.f;
}

__device__ inline void atomicMaxF(float* addr, float v) {
    int* ai = (int*)addr;
    int old = __float_as_int(*addr);
    while (__int_as_float(old) < v) {
        const int assumed = old;
        old = atomicCAS(ai, assumed, __float_as_int(v));
        if (old == assumed) break;
    }
}

// ---------------------------------------------------------------------------
// Fused 128->1024 conv + per-batch global max. Wave handles (mtile, batch,
// col-chunk); two interleaved column tiles (2 independent acc chains),
// running per-row max -> shuffle reduce -> CAS-max into g[1024x16].
// Avoids materializing the 128 MB [1024 x 32768] tensor.
// ---------------------------------------------------------------------------
__global__ void gemm_c2_max_kernel(const float* __restrict__ W,   // [1024,128]
                                   const float* __restrict__ X,   // [128, P]
                                   const float* __restrict__ bias,
                                   float* __restrict__ g) {       // [1024,16]
    const int MT = 64, NB = 8, NCH = 8, TPC = NPTS / 16 / NCH;    // 32 tiles
    const int wpb  = blockDim.x >> 5;
    const int wave = blockIdx.x * wpb + (threadIdx.x >> 5);
    if (wave >= MT * NB * NCH) return;
    const int mt  = wave / (NB * NCH);
    const int rem = wave - mt * (NB * NCH);
    const int b   = rem / NCH;
    const int ch  = rem - b * NCH;

    const int lane = threadIdx.x & 31;
    const int half = lane >> 4;
    const int l    = lane & 15;
    const int m0   = mt << 4;

    const float* wrow = W + (size_t)(m0 + l) * 128;

    float rmax[8];
#pragma unroll
    for (int r = 0; r < 8; ++r) rmax[r] = -__builtin_inff();

    for (int t = 0; t < TPC; t += 2) {
        const int n0 = b * NPTS + (ch * TPC + t) * 16;
        const float* xcol = X + n0 + l;
        v8f acc0 = {}, acc1 = {};
        for (int k0 = 0; k0 < 128; k0 += 4) {
            const int ka = k0 + 2 * half;
            v2f a, b0, b1;
            a.x  = wrow[ka];
            a.y  = wrow[ka + 1];
            const float* xr0 = xcol + (size_t)ka * PTOT;
            const float* xr1 = xcol + (size_t)(ka + 1) * PTOT;
            b0.x = xr0[0];  b0.y = xr1[0];
            b1.x = xr0[16]; b1.y = xr1[16];
            acc0 = __builtin_amdgcn_wmma_f32_16x16x4_f32(
                false, a, false, b0, (short)0, acc0, false, false);
            acc1 = __builtin_amdgcn_wmma_f32_16x16x4_f32(
                false, a, false, b1, (short)0, acc1, false, false);
        }
#pragma unroll
        for (int r = 0; r < 8; ++r)
            rmax[r] = fmaxf(rmax[r], fmaxf(acc0[r], acc1[r]));
    }
    // reduce across the 16 lanes of each half (same row set per half)
#pragma unroll
    for (int off = 8; off; off >>= 1)
#pragma unroll
        for (int r = 0; r < 8; ++r)
            rmax[r] = fmaxf(rmax[r], __shfl_xor(rmax[r], off, 32));

    if (l == 0) {
#pragma unroll
        for (int r = 0; r < 8; ++r) {
            const int row = m0 + r + 8 * half;
            atomicMaxF(&g[row * 16 + b], rmax[r] + bias[row]);
        }
    }
}

// out[b*512 + o] = h2[o*16 + b]
__global__ void final_copy_kernel(const float* __restrict__ h2,
                                  float* __restrict__ out) {
    const int t = blockIdx.x * blockDim.x + threadIdx.x;
    if (t >= BATCH * 512) return;
    const int b = t >> 9, o = t & 511;
    out[t] = h2[o * 16 + b];
}

// ---------------------------------------------------------------------------
extern "C" void kernel_launch(void* const* d_in, const int* in_sizes, int n_in,
                              void* d_out, int out_size, void* d_ws, size_t ws_size,
                              hipStream_t stream) {
    const float* pts  = (const float*)d_in[0];
    const float* m1w1 = (const float*)d_in[1];  const float* m1b1 = (const float*)d_in[2];
    const float* m1w2 = (const float*)d_in[3];  const float* m1b2 = (const float*)d_in[4];
    const float* m1w3 = (const float*)d_in[5];  const float* m1b3 = (const float*)d_in[6];
    const float* l1w  = (const float*)d_in[7];  const float* l1b  = (const float*)d_in[8];
    const float* c1w  = (const float*)d_in[9];  const float* c1b  = (const float*)d_in[10];
    const float* l2w  = (const float*)d_in[11]; const float* l2b  = (const float*)d_in[12];
    const float* c2w  = (const float*)d_in[13]; const float* c2b  = (const float*)d_in[14];
    const float* m2w1 = (const float*)d_in[15]; const float* m2b1 = (const float*)d_in[16];
    const float* m2w2 = (const float*)d_in[17]; const float* m2b2 = (const float*)d_in[18];
    float* out = (float*)d_out;

    // workspace carve-up
    char* ws = (char*)d_ws;
    int*   idx  = (int*)ws;                                   ws += (size_t)PTOT * KNN_K * sizeof(int);
    float* bufA = (float*)ws;                                 ws += (size_t)128 * PTOT * sizeof(float);
    float* bufB = (float*)ws;                                 ws += (size_t)128 * PTOT * sizeof(float);
    float* g    = (float*)ws;                                 ws += (size_t)1024 * 16 * sizeof(float);
    float* h1   = (float*)ws;                                 ws += (size_t)512 * 16 * sizeof(float);
    float* h2   = (float*)ws;

    // big GEMMs: 4 column tiles per wave
    auto gemm4 = [&](const float* W, const float* X, const float* bias, float* Y,
                     int Cout, int Cin, bool relu) {
        const int tiles = (Cout / 16) * (PTOT / 16 / 4);
        const int threads = 256, wpb = threads / 32;
        const int blocks = (tiles + wpb - 1) / wpb;
        if (relu)
            gemm_wmma_kernel<true, 4><<<blocks, threads, 0, stream>>>(W, X, bias, Y, Cout, Cin, PTOT);
        else
            gemm_wmma_kernel<false, 4><<<blocks, threads, 0, stream>>>(W, X, bias, Y, Cout, Cin, PTOT);
    };
    // mlp2 tail: single 16-wide column group
    auto gemm1 = [&](const float* W, const float* X, const float* bias, float* Y,
                     int Cout, int Cin, bool relu) {
        const int tiles = Cout / 16;
        const int threads = 256, wpb = threads / 32;
        const int blocks = (tiles + wpb - 1) / wpb;
        if (relu)
            gemm_wmma_kernel<true, 1><<<blocks, threads, 0, stream>>>(W, X, bias, Y, Cout, Cin, 16);
        else
            gemm_wmma_kernel<false, 1><<<blocks, threads, 0, stream>>>(W, X, bias, Y, Cout, Cin, 16);
    };

    // 1) KNN indices
    knn_kernel<<<dim3(NPTS / 256, BATCH), 256, 0, stream>>>(pts, idx);
    // 2) local covariance features [12, P] -> bufA
    local_cov_kernel<<<(PTOT + 255) / 256, 256, 0, stream>>>(pts, idx, bufA);
    // 3) mlp1
    gemm4(m1w1, bufA, m1b1, bufB, 64, 12, true);
    gemm4(m1w2, bufB, m1b2, bufA, 64, 64, true);
    gemm4(m1w3, bufA, m1b3, bufB, 64, 64, true);          // x3 -> bufB
    // 4) graph layer 1
    maxpool_kernel<<<(PTOT + 255) / 256, 256, 0, stream>>>(bufB, idx, bufA, 64);
    gemm4(l1w, bufA, l1b, bufB, 64, 64, false);           // t1 -> bufB
    gemm4(c1w, bufB, c1b, bufA, 128, 64, true);           // x4 -> bufA
    // 5) graph layer 2
    maxpool_kernel<<<(PTOT + 255) / 256, 256, 0, stream>>>(bufA, idx, bufB, 128);
    gemm4(l2w, bufB, l2b, bufA, 128, 128, false);         // t2 -> bufA
    // 6) fused 128->1024 conv + per-batch global max -> g [1024,16]
    init_g_kernel<<<(1024 * 16 + 255) / 256, 256, 0, stream>>>(g);
    {
        const int waves = 64 * 8 * 8, threads = 256, wpb = threads / 32;
        gemm_c2_max_kernel<<<(waves + wpb - 1) / wpb, threads, 0, stream>>>(c2w, bufA, c2b, g);
    }
    // 7) mlp2 on [*, 16] padded columns (cols 8..15 are zero pad)
    gemm1(m2w1, g, m2b1, h1, 512, 1024, true);
    gemm1(m2w2, h1, m2b2, h2, 512, 512, false);
    // 8) transpose to [B, 512]
    final_copy_kernel<<<(BATCH * 512 + 255) / 256, 256, 0, stream>>>(h2, out);
}